// GCN_LPA_1168231104601
// MI455X (gfx1250) — compile-verified
//
#include <hip/hip_runtime.h>

// Problem constants (match reference)
#define NNODES 50000
#define NEDGES 800000
#define F_IN   256
#define F_H    128
#define F_C    64
#define LPA_ITERS 10

typedef __attribute__((ext_vector_type(2))) float v2f;
typedef __attribute__((ext_vector_type(8))) float v8f;

// ---------------------------------------------------------------------------
// Degree accumulation: deg[row[e]] += edge_attr[e]
// ---------------------------------------------------------------------------
__global__ void k_deg(const int* __restrict__ ei, const float* __restrict__ ea,
                      float* __restrict__ deg, int E) {
    int e = blockIdx.x * blockDim.x + threadIdx.x;
    if (e < E) unsafeAtomicAdd(&deg[ei[e]], ea[e]);
}

// val[e] = ea[e] / deg[row[e]]  (0 if deg == 0)
__global__ void k_val(const int* __restrict__ ei, const float* __restrict__ ea,
                      const float* __restrict__ deg, float* __restrict__ val, int E) {
    int e = blockIdx.x * blockDim.x + threadIdx.x;
    if (e < E) {
        float d = deg[ei[e]];
        val[e] = (d > 0.0f) ? (ea[e] / d) : 0.0f;
    }
}

// out[i] = bias[i % C]  (C power of two)
__global__ void k_init_bias(const float* __restrict__ bias, float* __restrict__ out,
                            long total, int cmask) {
    long i = (long)blockIdx.x * blockDim.x + threadIdx.x;
    if (i < total) out[i] = bias[(int)(i & cmask)];
}

__global__ void k_relu(float* __restrict__ h, long total) {
    long i = (long)blockIdx.x * blockDim.x + threadIdx.x;
    if (i < total) h[i] = fmaxf(h[i], 0.0f);
}

// ---------------------------------------------------------------------------
// Scatter spmm: out[row[e], :] += val[e] * in[col[e], :]
// One thread handles one float4 quad of the feature row. C = 4<<LOGQ.
// ---------------------------------------------------------------------------
template <int LOGQ>
__global__ void k_scatter(const int* __restrict__ ei, const float* __restrict__ val,
                          const float* __restrict__ in, float* __restrict__ out, int E) {
    const int Q = 1 << LOGQ;        // quads per row
    const int C = Q * 4;            // feature width
    long tid = (long)blockIdx.x * blockDim.x + threadIdx.x;
    if (tid >= ((long)E << LOGQ)) return;
    int e = (int)(tid >> LOGQ);
    int q = (int)(tid & (Q - 1));
    int r = ei[e];          // destination row
    int c = ei[E + e];      // source row
    float v = val[e];
    float4 s = ((const float4*)(in + (size_t)c * C))[q];
    float* dst = out + (size_t)r * C + (size_t)q * 4;
    unsafeAtomicAdd(dst + 0, v * s.x);
    unsafeAtomicAdd(dst + 1, v * s.y);
    unsafeAtomicAdd(dst + 2, v * s.z);
    unsafeAtomicAdd(dst + 3, v * s.w);
}

// ---------------------------------------------------------------------------
// WMMA fp32 GEMM: C[M x NOUT] = A[M x K] @ B[K x NOUT]
// 256 threads = 8 waves/block. Each wave owns one 16x16 output tile.
// MB m-tiles per block, NT = NOUT/16 n-tiles; MB*NT == 8.
// A strip (MB*16 x K) and the FULL B matrix (K x NOUT) are staged in LDS
// (GEMM1: 144 KB dynamic LDS -- CDNA5 320KB WGP LDS only).
// Uses V_WMMA_F32_16X16X4_F32, K-loop of K/4 steps.
// ---------------------------------------------------------------------------
template <int K, int NOUT, int MB>
__global__ __launch_bounds__(256) void k_gemm(const float* __restrict__ A,
                                              const float* __restrict__ B,
                                              float* __restrict__ Cout, int M) {
    const int NT = NOUT / 16;                  // n tiles per block row
    extern __shared__ float smem[];
    float* sA = smem;                          // [MB*16][K]
    float* sB = smem + MB * 16 * K;            // [K][NOUT]

    const int tid  = threadIdx.x;
    const int row0 = blockIdx.x * (MB * 16);

    // Stage A strip (guard rows >= M with zeros)
    const int aQuads = MB * 16 * K / 4;
    for (int i = tid; i < aQuads; i += 256) {
        int row = row0 + (i * 4) / K;
        float4 v = make_float4(0.f, 0.f, 0.f, 0.f);
        if (row < M) v = ((const float4*)(A + (size_t)row0 * K))[i];
        ((float4*)sA)[i] = v;
    }
    // Stage full B matrix
    const int bQuads = K * NOUT / 4;
    for (int i = tid; i < bQuads; i += 256) {
        ((float4*)sB)[i] = ((const float4*)B)[i];
    }
    __syncthreads();

    const int w    = tid >> 5;       // wave id (wave32)
    const int lane = tid & 31;
    const int mt   = w / NT;
    const int nt   = w % NT;
    const int mTileG = blockIdx.x * MB + mt;
    if (mTileG * 16 >= M) return;    // wave-uniform guard (EXEC stays all-1s)

    const int l15 = lane & 15;
    const int hi  = lane >> 4;       // 0: K pair {0,1} / rows 0..7 ; 1: {2,3} / rows 8..15
    const int kA  = hi * 2;

    const float* aRow = sA + (mt * 16 + l15) * K;   // A row for this lane
    const float* bCol = sB + nt * 16 + l15;         // B column for this lane

    v8f acc = {};
#pragma unroll 4
    for (int k = 0; k < K; k += 4) {
        v2f a;
        a.x = aRow[k + kA];
        a.y = aRow[k + kA + 1];
        v2f b;
        b.x = bCol[(k + kA) * NOUT];
        b.y = bCol[(k + kA + 1) * NOUT];
        // (neg_a, A, neg_b, B, c_mod, C, reuse_a, reuse_b)
        acc = __builtin_amdgcn_wmma_f32_16x16x4_f32(false, a, false, b,
                                                    (short)0, acc, false, false);
    }

    // C/D layout: VGPR r -> row (r + hi*8), col = l15 (within tile)
    float* outp = Cout + (size_t)(mTileG * 16 + hi * 8) * NOUT + nt * 16 + l15;
#pragma unroll
    for (int r = 0; r < 8; ++r) outp[(size_t)r * NOUT] = acc[r];
}

// ---------------------------------------------------------------------------
// Host-side launch
// ---------------------------------------------------------------------------
static inline int cdiv(long a, int b) { return (int)((a + b - 1) / b); }

extern "C" void kernel_launch(void* const* d_in, const int* in_sizes, int n_in,
                              void* d_out, int out_size, void* d_ws, size_t ws_size,
                              hipStream_t stream) {
    const float* x    = (const float*)d_in[0];   // [N, 256]
    const float* soft = (const float*)d_in[1];   // [N, 64]
    const int*   ei   = (const int*)d_in[2];     // [2, E] (row then col), int32 per harness
    const float* ea   = (const float*)d_in[3];   // [E]
    const float* W1   = (const float*)d_in[4];   // [256, 128]
    const float* b1   = (const float*)d_in[5];   // [128]
    const float* W2   = (const float*)d_in[6];   // [128, 64]
    const float* b2   = (const float*)d_in[7];   // [64]
    float* out = (float*)d_out;                  // [N*64] out, then [N*64] labels

    const int N = NNODES, E = NEDGES;
    const long NC = (long)N * F_C;
    const long NH = (long)N * F_H;

    // Workspace carve-up (peak ~55 MB)
    char* ws = (char*)d_ws;
    size_t off = 0;
    auto carve = [&](size_t bytes) -> float* {
        float* p = (float*)(ws + off);
        off += (bytes + 255) & ~(size_t)255;
        return p;
    };
    float* deg  = carve((size_t)N * 4);          // [N]
    float* val  = carve((size_t)E * 4);          // [E]
    float* bufX = carve((size_t)NH * 4);         // xw [N,128] -> hw [N,64] -> labA [N,64]
    float* bufH = carve((size_t)NH * 4);         // h  [N,128] -> labB [N,64]

    const int T = 256;

    // 1) Row normalization: deg, then val = ea / deg[row]
    hipMemsetAsync(deg, 0, (size_t)N * 4, stream);
    k_deg<<<cdiv(E, T), T, 0, stream>>>(ei, ea, deg, E);
    k_val<<<cdiv(E, T), T, 0, stream>>>(ei, ea, deg, val, E);

    // 2) xw = x @ W1   (WMMA, 144KB LDS: 16x256 A-strip + full 256x128 W1)
    {
        const size_t lds = (size_t)(16 * F_IN + F_IN * F_H) * 4;   // 147456 B
        k_gemm<F_IN, F_H, 1><<<N / 16, T, lds, stream>>>(x, W1, bufX, N);
    }

    // 3) h = relu(spmm(xw) + b1): init with bias, scatter-add, relu
    k_init_bias<<<cdiv(NH, T), T, 0, stream>>>(b1, bufH, NH, F_H - 1);
    k_scatter<5><<<cdiv((long)E << 5, T), T, 0, stream>>>(ei, val, bufX, bufH, E);
    k_relu<<<cdiv(NH, T), T, 0, stream>>>(bufH, NH);

    // 4) hw = h @ W2   (WMMA, 48KB LDS)
    {
        const size_t lds = (size_t)(32 * F_H + F_H * F_C) * 4;     // 49152 B
        k_gemm<F_H, F_C, 2><<<cdiv(N, 32), T, lds, stream>>>(bufH, W2, bufX, N);
    }

    // 5) out = spmm(hw) + b2: init d_out[0:NC] with b2, scatter-add
    k_init_bias<<<cdiv(NC, T), T, 0, stream>>>(b2, out, NC, F_C - 1);
    k_scatter<4><<<cdiv((long)E << 4, T), T, 0, stream>>>(ei, val, bufX, out, E);

    // 6) Label propagation: 10 spmm steps, ping-pong; last step -> d_out[NC:2NC]
    float* labA = bufX;   // reuse (first N*64 floats of each buffer)
    float* labB = bufH;
    const float* src = soft;
    for (int it = 0; it < LPA_ITERS; ++it) {
        float* dst = (it == LPA_ITERS - 1) ? (out + NC) : ((it & 1) ? labB : labA);
        hipMemsetAsync(dst, 0, (size_t)NC * 4, stream);
        k_scatter<4><<<cdiv((long)E << 4, T), T, 0, stream>>>(ei, val, src, dst, E);
        src = dst;
    }
}